// ConvCaps_30966714204523
// MI455X (gfx1250) — compile-verified
//
#include <hip/hip_runtime.h>

typedef __attribute__((ext_vector_type(2))) float v2f;
typedef __attribute__((ext_vector_type(8))) float v8f;

#define CP    32
#define CN    32
#define Hh    18
#define Ww    18
#define Ho    16
#define Wo    16
#define NBAT  16
#define NTOT  (NBAT*Ho*Wo)   /* 4096 */
#define NPB   4              /* positions per block -> 16 WMMA columns */
#define NITERS 3

// LDS layout (floats):
//  s_pose    [NPB][CP][16]      2048
//  s_act     [NPB][CP]           128
//  s_vote    [NPB][CN][CP][16] 65536   (256 KB)
//  s_routing [NPB][CN][CP]      4096
//  s_r       [NPB][CN][CP]      4096   (reused as "ap")
//  s_mean    [NPB][CN][16]      2048
//  s_var     [NPB][CN][16]      2048
//  s_sumcost [NPB][CN]           128
// total = 80128 floats = 320512 bytes  (<= 320 KB WGP LDS)
#define SMEM_FLOATS (NPB*CP*16 + NPB*CP + NPB*CN*CP*16 + NPB*CN*CP + \
                     NPB*CN*CP + NPB*CN*16 + NPB*CN*16 + NPB*CN)

__global__ __launch_bounds__(256)
void convcaps_em_kernel(const float* __restrict__ inputPose,
                        const float* __restrict__ inputAct,
                        const float* __restrict__ Wg,
                        const float* __restrict__ Bv,
                        const float* __restrict__ Ba,
                        const float* __restrict__ invTemp_p,
                        const float* __restrict__ pi_p,
                        float* __restrict__ out)
{
    extern __shared__ float smem[];
    float* s_pose    = smem;                    // [NPB][CP][16]
    float* s_act     = s_pose    + NPB*CP*16;   // [NPB][CP]
    float* s_vote    = s_act     + NPB*CP;      // [NPB][CN][CP][16]
    float* s_routing = s_vote    + NPB*CN*CP*16;// [NPB][CN][CP]
    float* s_r       = s_routing + NPB*CN*CP;   // [NPB][CN][CP] (also ap)
    float* s_mean    = s_r       + NPB*CN*CP;   // [NPB][CN][16]
    float* s_var     = s_mean    + NPB*CN*16;   // [NPB][CN][16]
    float* s_sumcost = s_var     + NPB*CN*16;   // [NPB][CN]

    const int tid   = threadIdx.x;
    const int nbase = blockIdx.x * NPB;

    // ---------- Phase 0: fused 3x3/1 avg-pool of pose + activation ----------
    for (int e = tid; e < NPB*CP*16; e += 256) {
        int nl = e >> 9;            // /512
        int p  = (e >> 4) & 31;
        int m  = e & 15;
        int n  = nbase + nl;
        int b  = n >> 8;            // /(Ho*Wo)
        int ho = (n >> 4) & 15;
        int wo = n & 15;
        float s = 0.f;
        #pragma unroll
        for (int dy = 0; dy < 3; ++dy)
            #pragma unroll
            for (int dx = 0; dx < 3; ++dx)
                s += inputPose[((((b*Hh + ho+dy)*Ww + (wo+dx))*CP + p)<<4) + m];
        s_pose[(nl*CP + p)*16 + m] = s * (1.f/9.f);
    }
    for (int e = tid; e < NPB*CP; e += 256) {
        int nl = e >> 5, p = e & 31;
        int n  = nbase + nl;
        int b = n >> 8, ho = (n >> 4) & 15, wo = n & 15;
        float s = 0.f;
        #pragma unroll
        for (int dy = 0; dy < 3; ++dy)
            #pragma unroll
            for (int dx = 0; dx < 3; ++dx)
                s += inputAct[((b*Hh + ho+dy)*Ww + (wo+dx))*CP + p];
        s_act[e] = s * (1.f/9.f);
    }
    for (int e = tid; e < NPB*CN*CP; e += 256)
        s_routing[e] = 1.f / (float)CN;
    __syncthreads();

    // ---------- Phase 1: vote via V_WMMA_F32_16X16X4_F32 ----------
    // Per p: A = weights stacked (rows = c*4+i, K = j), 16-row tile per wave;
    //        B = pose stacked (K = j, cols = n_local*4 + k).
    {
        const int wave = tid >> 5;
        const int lane = tid & 31;
        const int col  = lane & 15;     // M for A-frag, N for B/D-frag
        const int half = lane >> 4;     // K-half selector
        const int cA   = wave*4 + (col >> 2);   // output capsule of this A row
        const int iA   = col & 3;               // row-within-4x4
        const int k0   = half*2;                // K base (j)
        const int nB   = col >> 2;              // n_local of this column
        const int kB   = col & 3;               // col-within-4x4
        for (int p = 0; p < CP; ++p) {
            // A-frag: W[cA][p][iA][k0], W[cA][p][iA][k0+1] (8B aligned)
            float2 aw = *(const float2*)(Wg + ((cA*CP + p)<<4) + iA*4 + k0);
            if (p + 1 < CP)
                __builtin_prefetch(Wg + ((cA*CP + p + 1)<<4), 0, 1);
            v2f a; a.x = aw.x; a.y = aw.y;
            // B-frag: pose[nB][p][j=k0][kB], pose[nB][p][j=k0+1][kB]
            v2f b;
            b.x = s_pose[(nB*CP + p)*16 + k0*4     + kB];
            b.y = s_pose[(nB*CP + p)*16 + (k0+1)*4 + kB];
            v8f acc = {};
            acc = __builtin_amdgcn_wmma_f32_16x16x4_f32(
                false, a, false, b, (short)0, acc, false, false);
            // D: VGPR v -> row v + 8*half within tile; N = col
            #pragma unroll
            for (int v = 0; v < 8; ++v) {
                int row = wave*16 + v + half*8;       // c*4 + i
                int cD = row >> 2, iD = row & 3;
                s_vote[((nB*CN + cD)*CP + p)*16 + iD*4 + kB] = acc[v];
            }
        }
    }
    __syncthreads();

    // ---------- Phase 2: EM routing (3 iterations, all in LDS) ----------
    float it = invTemp_p[0];
    const float two_pi = 2.0f * pi_p[0];
    const int g  = tid;
    const int nl = g >> 5;
    const int cc = g & 31;
    float aout = 0.f;

    for (int iter = 0; iter < NITERS; ++iter) {
        float sc = 0.f;
        if (g < NPB*CN) {
            // r = routing * a_in ; r_sum
            float rsum = 0.f;
            for (int p = 0; p < CP; ++p) {
                float r = s_routing[(nl*CN + cc)*CP + p] * s_act[nl*CP + p];
                s_r[(nl*CN + cc)*CP + p] = r;
                rsum += r;
            }
            // mean
            for (int m = 0; m < 16; ++m) {
                float acc = 0.f;
                for (int p = 0; p < CP; ++p)
                    acc += s_r[(nl*CN + cc)*CP + p] *
                           s_vote[((nl*CN + cc)*CP + p)*16 + m];
                s_mean[(nl*CN + cc)*16 + m] = acc / (rsum + 1e-5f);
            }
            // var
            for (int m = 0; m < 16; ++m) {
                float mu = s_mean[(nl*CN + cc)*16 + m];
                float acc = 0.f;
                for (int p = 0; p < CP; ++p) {
                    float d = s_vote[((nl*CN + cc)*CP + p)*16 + m] - mu;
                    acc += d*d * s_r[(nl*CN + cc)*CP + p];
                }
                s_var[(nl*CN + cc)*16 + m] = acc / (rsum + 0.001f);
            }
            // cost, sum over m
            for (int m = 0; m < 16; ++m) {
                float var = s_var[(nl*CN + cc)*16 + m];
                sc += (Bv[cc*16 + m] + logf(sqrtf(var + 1e-6f) + 1e-5f)) * rsum;
            }
            s_sumcost[nl*CN + cc] = sc;
        }
        __syncthreads();
        if (g < NPB*CN) {
            // cross-c stats -> act_out
            float scm = 0.f;
            for (int c2 = 0; c2 < CN; ++c2) scm += s_sumcost[nl*CN + c2];
            scm *= (1.f/(float)CN);
            float sv = 0.f;
            for (int c2 = 0; c2 < CN; ++c2) {
                float d = s_sumcost[nl*CN + c2] - scm;
                sv += d*d;
            }
            float stdv = sqrtf(sv*(1.f/(float)CN) + 1e-5f);
            float z = it * (Ba[cc] + (scm - sc) / (stdv + 1e-5f));
            aout = 1.f / (1.f + expf(-z));
            // prob, ap (reuse s_r row owned by this thread)
            float sumvar = 0.f;
            for (int m = 0; m < 16; ++m) sumvar += s_var[(nl*CN + cc)*16 + m];
            float denom = sqrtf(two_pi * sumvar) + 1e-4f;
            for (int p = 0; p < CP; ++p) {
                float e2 = 0.f;
                for (int m = 0; m < 16; ++m) {
                    float d = s_vote[((nl*CN + cc)*CP + p)*16 + m] -
                              s_mean[(nl*CN + cc)*16 + m];
                    e2 += d*d / (2.f*s_var[(nl*CN + cc)*16 + m] + 1e-5f);
                }
                s_r[(nl*CN + cc)*CP + p] = aout * (expf(-e2) / denom);
            }
        }
        __syncthreads();
        if (g < NPB*CP) {    // (n_local, p) mapping for routing renorm
            int pn = g >> 5, pp = g & 31;
            float aps = 0.f;
            for (int c2 = 0; c2 < CN; ++c2) aps += s_r[(pn*CN + c2)*CP + pp];
            float inv = 1.f / (aps + 1e-5f);
            for (int c2 = 0; c2 < CN; ++c2)
                s_routing[(pn*CN + c2)*CP + pp] = s_r[(pn*CN + c2)*CP + pp] * inv;
        }
        it += 1.0f;
        __syncthreads();
    }

    // ---------- Output: mean (N,CN,16) ++ act_out (N,CN) ----------
    for (int e = tid; e < NPB*CN*16; e += 256) {
        int nl2  = e >> 9;           // /(CN*16)
        int rest = e & 511;
        out[(size_t)(nbase + nl2)*CN*16 + rest] = s_mean[e];
    }
    if (g < NPB*CN)
        out[(size_t)NTOT*CN*16 + (size_t)(nbase + nl)*CN + cc] = aout;
}

extern "C" void kernel_launch(void* const* d_in, const int* in_sizes, int n_in,
                              void* d_out, int out_size, void* d_ws, size_t ws_size,
                              hipStream_t stream) {
    (void)in_sizes; (void)n_in; (void)d_ws; (void)ws_size; (void)out_size;
    const float* inputPose = (const float*)d_in[0];
    const float* inputAct  = (const float*)d_in[1];
    const float* Wg        = (const float*)d_in[2];
    const float* Bv        = (const float*)d_in[3];
    const float* Ba        = (const float*)d_in[4];
    const float* invT      = (const float*)d_in[8];
    const float* pi        = (const float*)d_in[9];
    float* out = (float*)d_out;

    const size_t shmem = (size_t)SMEM_FLOATS * sizeof(float);  // 320512 B
    hipLaunchKernelGGL(convcaps_em_kernel, dim3(NTOT/NPB), dim3(256), shmem,
                       stream, inputPose, inputAct, Wg, Bv, Ba, invT, pi, out);
}